// HungarianLoss_9629316677891
// MI455X (gfx1250) — compile-verified
//
#include <hip/hip_runtime.h>

#define N_DIM 256

typedef __attribute__((ext_vector_type(2))) float v2f;
typedef __attribute__((ext_vector_type(8))) float v8f;
typedef __attribute__((ext_vector_type(4))) int   v4i;

// Monotone mapping: float -> u32 such that key order == float order.
__device__ __forceinline__ unsigned fkey(float f) {
  unsigned b = __float_as_uint(f);
  return (b & 0x80000000u) ? ~b : (b | 0x80000000u);
}
__device__ __forceinline__ float funkey(unsigned k) {
  unsigned b = (k & 0x80000000u) ? (k & 0x7fffffffu) : ~k;
  return __uint_as_float(b);
}

__device__ __forceinline__ unsigned long long wave_min_u64(unsigned long long x) {
#pragma unroll
  for (int off = 16; off > 0; off >>= 1) {
    unsigned long long o = __shfl_xor(x, off, 32);
    if (o < x) x = o;
  }
  return x;
}

// One workgroup (256 threads = 8 wave32) solves one 256x256 assignment.
// Entire cost matrix lives in LDS (256KB of the WGP's 320KB).
__global__ __launch_bounds__(256) void hungarian_kernel(const float* __restrict__ D,
                                                        float* __restrict__ ws) {
  extern __shared__ float Dsh[];                 // N*N floats (dynamic, 256KB)
  __shared__ float u_[N_DIM + 1];                // row potentials (cross-thread)
  __shared__ int   p_[N_DIM + 1];                // p_[j] = row matched to col j (1-based)
  __shared__ int   way[N_DIM + 1];               // alternating-path predecessor
  __shared__ unsigned long long sMin[2];         // double-buffered argmin slot
  __shared__ float sval[N_DIM];
  __shared__ float wsum[8];

  const int tid  = threadIdx.x;
  const int lane = tid & 31;
  const int wid  = tid >> 5;
  const int b    = blockIdx.x;
  const float* Db = D + (size_t)b * N_DIM * N_DIM;

  // Kick the cachelines toward us while we set up (gfx1250 global_prefetch_b8).
  __builtin_prefetch(Db + (size_t)tid * N_DIM, 0, 3);

  // ---- Stage the 256KB cost matrix into LDS ----
#if __has_builtin(__builtin_amdgcn_global_load_async_to_lds_b128)
  {
    typedef v4i __attribute__((address_space(1)))* gv4p;   // global int4*
    typedef v4i __attribute__((address_space(3)))* lv4p;   // LDS int4*
    gv4p gsrc = (gv4p)Db;
    lv4p ldst = (lv4p)Dsh;
#pragma unroll 4
    for (int k = 0; k < 64; ++k) {               // 64 * 256 lanes * 16B = 256KB
      const int idx = k * 256 + tid;
      __builtin_amdgcn_global_load_async_to_lds_b128(gsrc + idx, ldst + idx, 0, 0);
    }
#if __has_builtin(__builtin_amdgcn_s_wait_asynccnt)
    __builtin_amdgcn_s_wait_asynccnt(0);
#else
    asm volatile("s_wait_asynccnt 0" ::: "memory");
#endif
  }
#else
  {
    const float4* src = (const float4*)Db;
    float4* dst = (float4*)Dsh;
#pragma unroll 4
    for (int k = 0; k < 64; ++k) dst[k * 256 + tid] = src[k * 256 + tid];
  }
#endif

  u_[tid] = 0.f; p_[tid] = 0;
  if (tid == 0) { u_[N_DIM] = 0.f; p_[N_DIM] = 0; }

  const int j = tid + 1;        // this thread owns column j for the whole kernel
  float vR = 0.f;               // column potential v[j] (persists across phases)
  __syncthreads();

  // ---- Jonker–Volgenant: one phase per row i ----
#pragma unroll 1
  for (int i = 1; i <= N_DIM; ++i) {
    float minvR = __builtin_inff();   // minv[j], thread-private
    bool  ju    = false;              // used[j], thread-private
    if (tid == 0) { p_[0] = i; sMin[0] = ~0ull; sMin[1] = ~0ull; }
    __syncthreads();

    int j0 = 0;
    int it = 0;
#pragma unroll 1
    for (;;) {
      const int   i0  = p_[j0];
      const float ui0 = u_[i0];
      if (!ju) {
        const float cur = Dsh[(i0 - 1) * N_DIM + tid] - ui0 - vR;
        if (cur < minvR) { minvR = cur; way[j] = j0; }
      }
      // 256-wide argmin over unused columns: wave32 shuffle min + one LDS u64 atomic.
      unsigned long long pk = ju ? ~0ull
          : ((((unsigned long long)fkey(minvR)) << 32) | (unsigned)j);
      pk = wave_min_u64(pk);
      const int par = it & 1;
      if (lane == 0) atomicMin(&sMin[par], pk);
      __syncthreads();                                  // barrier 1

      const unsigned long long res = sMin[par];
      const int   j1    = (int)(res & 0xffffffffull);
      const float delta = funkey((unsigned)(res >> 32));

      // Potential updates (assigned rows of used cols are distinct -> no conflicts).
      if (ju) { vR -= delta; u_[p_[j]] += delta; }
      else    { minvR -= delta; }
      if (tid == 0) { u_[p_[0]] += delta; sMin[par ^ 1] = ~0ull; }  // re-arm other slot
      __syncthreads();                                  // barrier 2

      if (j == j1) ju = true;   // j1 joins the used set for the NEXT iteration
      j0 = j1; ++it;
      if (p_[j0] == 0) break;   // reached an unmatched column
    }

    // Augment along the alternating path (short, serial on one thread).
    if (tid == 0) {
      int jj = j0;
      while (jj) { const int jp = way[jj]; p_[jj] = p_[jp]; jj = jp; }
    }
    __syncthreads();
  }

  // ---- Gather matched costs: column j is matched to row p_[j]-1 ----
  sval[tid] = Dsh[(p_[j] - 1) * N_DIM + tid];
  __syncthreads();

  // ---- Sum the 256 matched costs on the matrix pipe ----
  float bsum = 0.f;
#if __has_builtin(__builtin_amdgcn_wmma_f32_16x16x4_f32)
  if (wid < 4) {                                 // wave-uniform branch: EXEC all ones
    v2f a;  a.x = sval[wid * 64 + lane];  a.y = sval[wid * 64 + 32 + lane];
    v2f bones; bones.x = 1.f; bones.y = 1.f;     // B = ones => D[m][n] = rowsum(A, m)
    v8f c = {};
    c = __builtin_amdgcn_wmma_f32_16x16x4_f32(false, a, false, bones, (short)0, c,
                                              false, false);
    // Per lane: sum of its 8 D rows. lane0 holds rows 0..7, lane16 rows 8..15;
    // their sum is exactly the total of this wave's 64 A values.
    const float part = c[0] + c[1] + c[2] + c[3] + c[4] + c[5] + c[6] + c[7];
    const float tot  = __shfl(part, 0, 32) + __shfl(part, 16, 32);
    if (lane == 0) wsum[wid] = tot;
  }
  __syncthreads();
  if (tid == 0) bsum = wsum[0] + wsum[1] + wsum[2] + wsum[3];
#else
  for (int k = 128; k > 0; k >>= 1) {
    if (tid < k) sval[tid] += sval[tid + k];
    __syncthreads();
  }
  bsum = sval[0];
#endif
  if (tid == 0) ws[b] = bsum;
}

// Deterministic fixed-order reduction of per-batch sums -> scalar mean.
__global__ __launch_bounds__(128) void finalize_kernel(const float* __restrict__ ws,
                                                       float* __restrict__ out,
                                                       int B, float scale) {
  __shared__ float s[128];
  const int t = threadIdx.x;
  float acc = 0.f;
  for (int k = t; k < B; k += 128) acc += ws[k];
  s[t] = acc;
  __syncthreads();
  for (int k = 64; k > 0; k >>= 1) {
    if (t < k) s[t] += s[t + k];
    __syncthreads();
  }
  if (t == 0) out[0] = s[0] * scale;
}

extern "C" void kernel_launch(void* const* d_in, const int* in_sizes, int n_in,
                              void* d_out, int out_size, void* d_ws, size_t ws_size,
                              hipStream_t stream) {
  (void)n_in; (void)out_size; (void)ws_size;
  const float* D  = (const float*)d_in[0];
  float* out      = (float*)d_out;
  float* ws       = (float*)d_ws;                // 1 float per batch
  const int B     = in_sizes[0] / (N_DIM * N_DIM);
  const size_t sh = (size_t)N_DIM * N_DIM * sizeof(float);  // 256KB LDS / WG

  hungarian_kernel<<<B, N_DIM, sh, stream>>>(D, ws);
  finalize_kernel<<<1, 128, 0, stream>>>(ws, out, B, 1.0f / (float)(B * N_DIM));
}